// NetFV_3642132267527
// MI455X (gfx1250) — compile-verified
//
#include <hip/hip_runtime.h>

typedef float v2f  __attribute__((ext_vector_type(2)));
typedef float v8f  __attribute__((ext_vector_type(8)));
typedef float f32x4 __attribute__((ext_vector_type(4)));

#define F_   60
#define M_   600
#define C_   8
#define OUT_ 18
#define TILE 64
#define XSTR 72                         // padded f-stride: half-waves hit disjoint LDS banks
#define NQ   (TILE * 15)                // float4 quads per tile (64 rows x 15)
#define NTILE ((M_ + TILE - 1) / TILE)  // 10

__launch_bounds__(256, 1)
__global__ void netfv_fused(const float* __restrict__ x,      // [B*M, F]
                            const float* __restrict__ cw,     // cluster_weights [F, C]
                            const float* __restrict__ covw,   // covar_weights [F, C]
                            const float* __restrict__ cbias,  // [C]
                            const float* __restrict__ cw2,    // cluster_weights2 [1, F, C]
                            const float* __restrict__ hw,     // hidden1_weights [2*C*F, OUT]
                            float* __restrict__ out)          // [B, OUT]
{
    __shared__ float Xlds[2][TILE * XSTR]; // double-buffered X tile, cols 60..71 zero
    __shared__ float Wlds[F_ * C_];
    __shared__ float Blds[C_];
    __shared__ float Alds[TILE * 16];      // act tile, cols 8..15 zero
    __shared__ float ASP[TILE * C_];       // a_sum partials
    __shared__ float ASUM[C_];
    __shared__ float FT[2 * TILE * C_];    // fv1/fv2 tiles [tensor][f][c], atomically summed
    __shared__ float FV1[F_ * C_];
    __shared__ float FV2[F_ * C_];
    __shared__ float NCL[C_];
    __shared__ float RED1[256];
    __shared__ float RED2[256];
    __shared__ float GN[2];
    __shared__ float FVV[2 * F_ * C_];

    const int t    = threadIdx.x;
    const int lane = t & 31;
    const int w    = t >> 5;            // wave id 0..7
    const int b    = blockIdx.x;

    // ---- one-time LDS init ----
    for (int i = t; i < F_ * C_; i += 256) Wlds[i] = cw[i];
    if (t < C_) Blds[t] = cbias[t];
    for (int i = t; i < TILE * 8; i += 256) {            // zero act cols 8..15
        int r = i >> 3, c = i & 7;
        Alds[r * 16 + 8 + c] = 0.f;
    }
    for (int i = t; i < TILE * (XSTR - F_); i += 256) {  // zero X pad cols 60..71 (both bufs)
        int r = i / (XSTR - F_), c = i % (XSTR - F_);
        Xlds[0][r * XSTR + F_ + c] = 0.f;
        Xlds[1][r * XSTR + F_ + c] = 0.f;
    }
    for (int i = t; i < 2 * TILE * C_; i += 256) FT[i] = 0.f;

    // wave roles: each wave computes BOTH tensors for one (fblock, K-half)
    const int fblock = w & 3;           // 16-row f block of the [60x8] output
    const int khalf  = w >> 2;          // first/second half of the 16 k-steps
    v8f acc1 = {};                       // fv1 accumulator
    v8f acc2 = {};                       // fv2 accumulator
    float asum_loc[C_];
#pragma unroll
    for (int c = 0; c < C_; ++c) asum_loc[c] = 0.f;

    const size_t xbase = (size_t)b * M_ * F_;

    // ---- register-staged tile loader (double buffer) ----
    f32x4 rg[4];
    auto issue_loads = [&](int tile_i) {
        const int m0 = tile_i * TILE;
#pragma unroll
        for (int j = 0; j < 4; ++j) {
            int idx = t + j * 256;
            f32x4 v = {};
            if (idx < NQ) {
                int r = idx / 15, q = idx % 15;
                if (m0 + r < M_)
                    v = *(const f32x4*)(x + xbase + (size_t)(m0 + r) * F_ + q * 4);
            }
            rg[j] = v;
        }
    };
    auto store_tile = [&](int buf) {
#pragma unroll
        for (int j = 0; j < 4; ++j) {
            int idx = t + j * 256;
            if (idx < NQ) {
                int r = idx / 15, q = idx % 15;
                *(f32x4*)&Xlds[buf][r * XSTR + q * 4] = rg[j];
            }
        }
    };

    // prologue: fill buffer 0
    issue_loads(0);
    store_tile(0);

    const int fl   = fblock * 16 + (lane & 15); // A row (f); f>=60 zero-padded
    const int koff = (lane >> 4) << 1;          // 0 or 2 (K pair per half-wave)
    const int nn   = lane & 15;                 // B col (c); cols >=8 zero

    // =========================== main streaming loop ===========================
    for (int tile = 0; tile < NTILE; ++tile) {
        const int cur = tile & 1;
        __syncthreads();                // buf[cur] + Alds ready; prior consumers done

        if (tile + 1 < NTILE) issue_loads(tile + 1);   // hide HBM latency behind compute

        // ---- logits: X @ W + bias (each thread: one row, two clusters) ----
        {
            int r  = t & 63;
            int c0 = t >> 6;                   // 0..3 -> clusters c0, c0+4
            float a0 = Blds[c0], a1 = Blds[c0 + 4];
#pragma unroll 4
            for (int k = 0; k < F_; ++k) {
                float xv = Xlds[cur][r * XSTR + k];
                a0 = fmaf(xv, Wlds[k * C_ + c0], a0);
                a1 = fmaf(xv, Wlds[k * C_ + c0 + 4], a1);
            }
            Alds[r * 16 + c0]     = a0;
            Alds[r * 16 + c0 + 4] = a1;
        }
        __syncthreads();

        // ---- softmax over C=8 per row; accumulate a_sum in registers ----
        if (t < TILE) {
            int r = t;
            if (tile * TILE + r < M_) {
                float l[C_];
                float mx = -1e30f;
#pragma unroll
                for (int c = 0; c < C_; ++c) { l[c] = Alds[r * 16 + c]; mx = fmaxf(mx, l[c]); }
                float s = 0.f;
#pragma unroll
                for (int c = 0; c < C_; ++c) { l[c] = __expf(l[c] - mx); s += l[c]; }
                float inv = 1.f / s;
#pragma unroll
                for (int c = 0; c < C_; ++c) {
                    float a = l[c] * inv;
                    Alds[r * 16 + c] = a;
                    asum_loc[c] += a;
                }
            } else {
#pragma unroll
                for (int c = 0; c < C_; ++c) Alds[r * 16 + c] = 0.f;
            }
        }
        __syncthreads();

        // ---- WMMA: both tensors per wave; one A/B fetch feeds two WMMAs ----
        {
#pragma unroll
            for (int ks = 0; ks < 8; ++ks) {
                int m = (khalf * 8 + ks) * 4 + koff;
                float ax = Xlds[cur][m * XSTR + fl];
                float ay = Xlds[cur][(m + 1) * XSTR + fl];
                float bx = Alds[m * 16 + nn];
                float by = Alds[(m + 1) * 16 + nn];
                v2f A  = {ax, ay};
                v2f Bv = {bx, by};
                acc1 = __builtin_amdgcn_wmma_f32_16x16x4_f32(
                         false, A, false, Bv, (short)0, acc1, false, false);
                v2f A2 = {ax * ax, ay * ay};
                acc2 = __builtin_amdgcn_wmma_f32_16x16x4_f32(
                         false, A2, false, Bv, (short)0, acc2, false, false);
            }
        }

        // spill next tile's registers into the other buffer (no conflict with cur)
        if (tile + 1 < NTILE) store_tile(1 - cur);
    }
    __syncthreads();

    // ---- merge K-half partials: lane l<16 -> rows j, lane>=16 -> rows j+8 ----
    {
        int n  = lane & 15;
        int fb = fblock * 16 + ((lane >> 4) << 3);
        if (n < C_) {
#pragma unroll
            for (int j = 0; j < 8; ++j) {
                atomicAdd(&FT[0 * TILE * C_ + (fb + j) * C_ + n], acc1[j]);
                atomicAdd(&FT[1 * TILE * C_ + (fb + j) * C_ + n], acc2[j]);
            }
        }
    }
    // ---- a_sum reduce ----
    if (t < TILE) {
#pragma unroll
        for (int c = 0; c < C_; ++c) ASP[t * C_ + c] = asum_loc[c];
    }
    __syncthreads();
    if (t < C_) {
        float s = 0.f;
        for (int i = 0; i < TILE; ++i) s += ASP[i * C_ + t];
        ASUM[t] = s;
    }
    __syncthreads();

    // ---- elementwise transforms ----
    for (int idx = t; idx < F_ * C_; idx += 256) {
        int f = idx >> 3, c = idx & 7;
        float asc = ASUM[c];
        float w2  = cw2[idx];
        float cv  = covw[idx];
        float cwv = cv * cv + 1e-6f;
        float r1  = FT[0 * TILE * C_ + f * C_ + c];
        float r2  = FT[1 * TILE * C_ + f * C_ + c];
        FV2[idx] = (asc * w2 * w2 + r2 - 2.f * r1 * w2) / (cwv * cwv) - asc;
        FV1[idx] = (r1 - asc * w2) / cwv;
    }
    __syncthreads();

    // ---- fv1: per-cluster L2 norm along F ----
    if (t < C_) {
        float s = 0.f;
        for (int f = 0; f < F_; ++f) { float y = FV1[f * C_ + t]; s += y * y; }
        NCL[t] = rsqrtf(fmaxf(s, 1e-12f));
    }
    __syncthreads();

    // ---- apply per-cluster scale; global sum-of-squares for both vectors ----
    {
        float s1 = 0.f, s2 = 0.f;
        for (int idx = t; idx < F_ * C_; idx += 256) {
            int c = idx & 7;
            float y = FV1[idx] * NCL[c];
            FV1[idx] = y;  s1 += y * y;
            float z = FV2[idx];  s2 += z * z;
        }
        RED1[t] = s1;  RED2[t] = s2;
    }
    __syncthreads();
    if (t < 32) {
        float a = 0.f, bb = 0.f;
        for (int j = t; j < 256; j += 32) { a += RED1[j]; bb += RED2[j]; }
#pragma unroll
        for (int off = 16; off > 0; off >>= 1) {
            a  += __shfl_down(a, off, 32);
            bb += __shfl_down(bb, off, 32);
        }
        if (t == 0) {
            GN[0] = rsqrtf(fmaxf(a,  1e-12f));
            GN[1] = rsqrtf(fmaxf(bb, 1e-12f));   // second l2_normalize is idempotent
        }
    }
    __syncthreads();
    for (int idx = t; idx < F_ * C_; idx += 256) {
        FVV[idx]           = FV1[idx] * GN[0];
        FVV[F_ * C_ + idx] = FV2[idx] * GN[1];
    }
    __syncthreads();

    // ---- out[b] = fv(960) @ hidden1_weights(960x18); weights L2-resident ----
    for (int o = w; o < OUT_; o += 8) {
        float s = 0.f;
        for (int i = lane; i < 2 * F_ * C_; i += 32)
            s = fmaf(FVV[i], hw[i * OUT_ + o], s);
#pragma unroll
        for (int off = 16; off > 0; off >>= 1) s += __shfl_down(s, off, 32);
        if (lane == 0) out[b * OUT_ + o] = s;
    }
}

extern "C" void kernel_launch(void* const* d_in, const int* in_sizes, int n_in,
                              void* d_out, int out_size, void* d_ws, size_t ws_size,
                              hipStream_t stream) {
    const float* x     = (const float*)d_in[0];   // reshaped_input [B*M, F]
    const float* cwp   = (const float*)d_in[1];   // cluster_weights [F, C]
    const float* covw  = (const float*)d_in[2];   // covar_weights [F, C]
    const float* cbias = (const float*)d_in[3];   // cluster_biases [C]
    const float* cw2   = (const float*)d_in[4];   // cluster_weights2 [1, F, C]
    const float* hw    = (const float*)d_in[5];   // hidden1_weights [2*C*F, OUT]
    float* out = (float*)d_out;

    const int B = in_sizes[0] / (M_ * F_);        // 2048
    netfv_fused<<<B, 256, 0, stream>>>(x, cwp, covw, cbias, cw2, hw, out);
}